// RoIHeadTemplate_781684047997
// MI455X (gfx1250) — compile-verified
//
#include <hip/hip_runtime.h>
#include <hip/hip_bf16.h>
#include <stdint.h>

#define B_      4
#define N_      16384
#define C_      3
#define PRE_    4096
#define POST_   512
#define THRESH_ 0.8f
#define TWO_PI_ 6.2831853071795864769f
#define PI_     3.1415926535897932385f

// ---- float <-> order-preserving u32 (ascending uint == ascending float) ----
__device__ __forceinline__ unsigned f2mono(float f) {
  unsigned b = __float_as_uint(f);
  return (b & 0x80000000u) ? ~b : (b | 0x80000000u);
}
__device__ __forceinline__ float mono2f(unsigned m) {
  unsigned b = (m & 0x80000000u) ? (m & 0x7FFFFFFFu) : ~m;
  return __uint_as_float(b);
}
__device__ __forceinline__ unsigned umin_(unsigned a, unsigned b) { return a < b ? a : b; }

// ---- CDNA5 async global -> LDS copy (ASYNCcnt-tracked) ----
__device__ __forceinline__ void async_copy_b32(void* lds_dst, const void* gsrc) {
  unsigned lo = (unsigned)(uintptr_t)lds_dst;                 // LDS byte offset (addr[31:0])
  unsigned long long ga = (unsigned long long)(uintptr_t)gsrc;
  asm volatile("global_load_async_to_lds_b32 %0, %1, off"
               :: "v"(lo), "v"(ga) : "memory");
}
__device__ __forceinline__ void wait_async0() {
#if defined(__has_builtin) && __has_builtin(__builtin_amdgcn_s_wait_asynccnt)
  __builtin_amdgcn_s_wait_asynccnt(0);
#else
  asm volatile("s_wait_asynccnt 0x0" ::: "memory");
#endif
}

// =====================================================================
// Kernel A: per-batch score/argmax, full in-LDS bitonic sort of 16384
// 64-bit keys (descending score, ascending index), emit top-4096.
// One workgroup per batch; 128 KB dynamic LDS (CDNA5: 320 KB/WGP).
// =====================================================================
__global__ void __launch_bounds__(1024)
topk_sort_kernel(const float* __restrict__ box, const float* __restrict__ cls,
                 unsigned* __restrict__ top_idx, float* __restrict__ top_score) {
  extern __shared__ unsigned long long keys[];  // N_ entries
  const int b   = blockIdx.x;
  const int tid = threadIdx.x;
  const int nthr = blockDim.x;

  for (int e = tid; e < N_; e += nthr) {
    const float* cp = cls + ((size_t)b * N_ + e) * C_;
    float m = fmaxf(cp[0], fmaxf(cp[1], cp[2]));
    // warm L2 for the NMS kernel's scattered box gather
    __builtin_prefetch(box + ((size_t)b * N_ + e) * 7, 0, 1);
    unsigned mono = f2mono(m);
    keys[e] = ((unsigned long long)(~mono) << 32) | (unsigned)e;
  }
  __syncthreads();

  // bitonic sort, ascending key == descending score (index breaks ties asc.)
  for (unsigned k = 2; k <= (unsigned)N_; k <<= 1) {
    for (unsigned j = k >> 1; j > 0; j >>= 1) {
      for (unsigned i = tid; i < (unsigned)N_; i += (unsigned)nthr) {
        unsigned ixj = i ^ j;
        if (ixj > i) {
          unsigned long long a = keys[i], c = keys[ixj];
          bool up = ((i & k) == 0);
          if ((a > c) == up) { keys[i] = c; keys[ixj] = a; }
        }
      }
      __syncthreads();
    }
  }

  for (int r = tid; r < PRE_; r += nthr) {
    unsigned long long kk = keys[r];
    top_idx[b * PRE_ + r]   = (unsigned)kk;
    top_score[b * PRE_ + r] = mono2f(~(unsigned)(kk >> 32));
  }
}

// =====================================================================
// Kernel B: greedy NMS. Geometry (x,y,dx,dy) of the 4096 candidates is
// gathered into LDS with async global->LDS loads; IoU rows are computed
// on the fly (never materializing the 4096x4096 matrix). First-valid
// index per round via wave32 shfl min-reduction (3 barriers/round).
// =====================================================================
__global__ void __launch_bounds__(1024)
nms_kernel(const float* __restrict__ box, const float* __restrict__ cls,
           const unsigned* __restrict__ top_idx, const float* __restrict__ top_score,
           float* __restrict__ rois, float* __restrict__ scores, float* __restrict__ labels) {
  extern __shared__ unsigned char smem[];
  float* bx  = (float*)smem;                       // PRE_
  float* by  = bx  + PRE_;                         // PRE_
  float* bdx = by  + PRE_;                         // PRE_
  float* bdy = bdx + PRE_;                         // PRE_
  unsigned char* valid = (unsigned char*)(bdy + PRE_);   // PRE_ bytes
  unsigned* selbuf = (unsigned*)(valid + PRE_);    // POST_
  unsigned* wred   = selbuf + POST_;               // 32
  unsigned* curI   = wred + 32;                    // 1

  const int b    = blockIdx.x;
  const int tid  = threadIdx.x;
  const int lane = tid & 31;
  const int wave = tid >> 5;

  // ---- stage candidate geometry: async gather into LDS ----
  for (int k = 0; k < 4; ++k) {
    int r = tid * 4 + k;
    unsigned idx = top_idx[b * PRE_ + r];
    const float* bp = box + ((size_t)b * N_ + idx) * 7;
    async_copy_b32(&bx[r],  bp + 0);
    async_copy_b32(&by[r],  bp + 1);
    async_copy_b32(&bdx[r], bp + 3);
    async_copy_b32(&bdy[r], bp + 4);
    valid[r] = 1;
  }
  wait_async0();
  __syncthreads();

  // ---- 512 sequential greedy rounds ----
  for (int t = 0; t < POST_; ++t) {
    unsigned lmin = 0xFFFFFFFFu;
    #pragma unroll
    for (int k = 0; k < 4; ++k) {
      int r = tid * 4 + k;
      if (lmin == 0xFFFFFFFFu && valid[r]) lmin = (unsigned)r;
    }
    #pragma unroll
    for (int off = 16; off > 0; off >>= 1)
      lmin = umin_(lmin, (unsigned)__shfl_xor((int)lmin, off, 32));
    if (lane == 0) wred[wave] = lmin;
    __syncthreads();
    if (tid < 32) {
      unsigned v = wred[tid];
      #pragma unroll
      for (int off = 16; off > 0; off >>= 1)
        v = umin_(v, (unsigned)__shfl_xor((int)v, off, 32));
      if (tid == 0) {
        *curI = v;
        selbuf[t] = (v == 0xFFFFFFFFu) ? 0u : (v | 0x80000000u); // bit31 = keep
      }
    }
    __syncthreads();

    unsigned i = *curI;
    if (i != 0xFFFFFFFFu) {
      float xi = bx[i], yi = by[i], dxi = bdx[i], dyi = bdy[i];
      float x1i = xi - 0.5f * dxi, x2i = xi + 0.5f * dxi;
      float y1i = yi - 0.5f * dyi, y2i = yi + 0.5f * dyi;
      float ai = dxi * dyi;
      #pragma unroll
      for (int k = 0; k < 4; ++k) {
        int r = tid * 4 + k;
        if (valid[r]) {
          float xr = bx[r], yr = by[r], dxr = bdx[r], dyr = bdy[r];
          float x1r = xr - 0.5f * dxr, x2r = xr + 0.5f * dxr;
          float y1r = yr - 0.5f * dyr, y2r = yr + 0.5f * dyr;
          float ix = fmaxf(fminf(x2i, x2r) - fmaxf(x1i, x1r), 0.0f);
          float iy = fmaxf(fminf(y2i, y2r) - fmaxf(y1i, y1r), 0.0f);
          float inter = ix * iy;
          float iou = inter / (ai + dxr * dyr - inter + 1e-6f);
          if (iou > THRESH_ || (unsigned)r == i) valid[r] = 0;
        }
      }
    }
    __syncthreads();
  }

  // ---- emit rois / scores / labels ----
  for (int t = tid; t < POST_; t += blockDim.x) {
    unsigned sb   = selbuf[t];
    bool keep     = (sb >> 31) != 0;
    unsigned sel  = sb & 0x7FFFFFFFu;
    unsigned src  = top_idx[b * PRE_ + sel];
    float kf      = keep ? 1.0f : 0.0f;
    const float* bp = box + ((size_t)b * N_ + src) * 7;
    float* ro = rois + ((size_t)b * POST_ + t) * 7;
    #pragma unroll
    for (int d = 0; d < 7; ++d) ro[d] = bp[d] * kf;
    scores[b * POST_ + t] = top_score[b * PRE_ + sel] * kf;
    const float* cp = cls + ((size_t)b * N_ + src) * C_;
    int lab = 0; float bv = cp[0];
    if (cp[1] > bv) { bv = cp[1]; lab = 1; }
    if (cp[2] > bv) { bv = cp[2]; lab = 2; }
    labels[b * POST_ + t] = keep ? (float)(lab + 1) : 1.0f;
  }
}

// =====================================================================
// Kernel C: canonical transform of gt_of_rois (per-row rotation + wrap)
// =====================================================================
__device__ __forceinline__ float fmodpos(float x, float m) {
  return x - floorf(x / m) * m;   // python-style mod (sign of divisor)
}

__global__ void __launch_bounds__(256)
canon_kernel(const float* __restrict__ rois, const float* __restrict__ gt,
             float* __restrict__ ct) {
  int g = blockIdx.x * blockDim.x + threadIdx.x;
  if (g >= B_ * POST_) return;
  const float* roi = rois + (size_t)g * 7;
  const float* gp  = gt   + (size_t)g * 8;
  float ry = fmodpos(roi[6], TWO_PI_);
  float x = gp[0] - roi[0], y = gp[1] - roi[1], z = gp[2] - roi[2];
  float heading = gp[6] - ry;
  float sa, ca;
  sincosf(-ry, &sa, &ca);
  float nx = x * ca - y * sa;
  float ny = x * sa + y * ca;
  float h = fmodpos(heading, TWO_PI_);
  if (h > 0.5f * PI_ && h < 1.5f * PI_) h = fmodpos(h + PI_, TWO_PI_);
  if (h > PI_) h -= TWO_PI_;
  h = fminf(fmaxf(h, -0.5f * PI_), 0.5f * PI_);
  float* o = ct + (size_t)g * 8;
  o[0] = nx;    o[1] = ny;    o[2] = z;     o[3] = gp[3];
  o[4] = gp[4]; o[5] = gp[5]; o[6] = h;     o[7] = gp[7];
}

extern "C" void kernel_launch(void* const* d_in, const int* in_sizes, int n_in,
                              void* d_out, int out_size, void* d_ws, size_t ws_size,
                              hipStream_t stream) {
  const float* box = (const float*)d_in[0];   // (B, N, 7)
  const float* cls = (const float*)d_in[1];   // (B, N, 3)
  const float* gt  = (const float*)d_in[2];   // (B, POST, 8)

  float* out    = (float*)d_out;
  float* rois   = out;                          // B*POST*7
  float* scores = rois   + B_ * POST_ * 7;      // B*POST
  float* labels = scores + B_ * POST_;          // B*POST
  float* ct     = labels + B_ * POST_;          // B*POST*8

  unsigned* top_idx  = (unsigned*)d_ws;                   // B*PRE u32
  float*    top_score = (float*)(top_idx + B_ * PRE_);    // B*PRE f32

  size_t ldsA = (size_t)N_ * sizeof(unsigned long long);  // 128 KB
  topk_sort_kernel<<<B_, 1024, ldsA, stream>>>(box, cls, top_idx, top_score);

  size_t ldsB = (size_t)4 * PRE_ * sizeof(float)   // bx,by,bdx,bdy
              + (size_t)PRE_                       // valid bytes
              + (size_t)POST_ * sizeof(unsigned)   // selbuf
              + 33 * sizeof(unsigned) + 64;        // wred + curI + pad
  nms_kernel<<<B_, 1024, ldsB, stream>>>(box, cls, top_idx, top_score,
                                         rois, scores, labels);

  canon_kernel<<<(B_ * POST_ + 255) / 256, 256, 0, stream>>>(rois, gt, ct);
}